// TR_GraphAttention_13417477833157
// MI455X (gfx1250) — compile-verified
//
#include <hip/hip_runtime.h>

#define NODE_F   128
#define ENT_F    256
#define OUT_STRIDE 768   // ENT_F * (DEPTH+1)

typedef __attribute__((ext_vector_type(2))) float v2f;
typedef __attribute__((ext_vector_type(8))) float v8f;

// ---- monotonic float <-> uint encoding for atomic max over signed floats ----
__device__ __forceinline__ unsigned enc_f32(float x) {
    unsigned b = __float_as_uint(x);
    return (b & 0x80000000u) ? ~b : (b | 0x80000000u);
}
__device__ __forceinline__ float dec_f32(unsigned e) {
    unsigned b = (e & 0x80000000u) ? (e & 0x7fffffffu) : ~e;
    return __uint_as_float(b);
}

// ---------------- generic zero ----------------
__global__ void k_zero(float* p, long long n) {
    long long i = (long long)blockIdx.x * blockDim.x + threadIdx.x;
    long long stride = (long long)gridDim.x * blockDim.x;
    for (; i < n; i += stride) p[i] = 0.0f;
}

// ---------------- degree count ----------------
__global__ void k_deg(const int* __restrict__ rows, int e, float* __restrict__ deg) {
    int i = blockIdx.x * blockDim.x + threadIdx.x;
    if (i < e) atomicAdd(deg + rows[i], 1.0f);
}

// ---------------- mean aggregation: one wave per edge, 128 floats ----------------
__global__ void k_mean_agg(const int* __restrict__ rows, const int* __restrict__ cols,
                           const float* __restrict__ emb, const float* __restrict__ deg,
                           float* __restrict__ outbase, int e) {
    long long t = (long long)blockIdx.x * blockDim.x + threadIdx.x;
    int edge = (int)(t >> 5);
    int lane = (int)t & 31;
    if (edge >= e) return;
    int r = rows[edge], c = cols[edge];
    float w = 1.0f / deg[r];
    const float4 v = *(const float4*)(emb + (size_t)c * NODE_F + lane * 4);
    float* dst = outbase + (size_t)r * OUT_STRIDE + lane * 4;
    atomicAdd(dst + 0, w * v.x);
    atomicAdd(dst + 1, w * v.y);
    atomicAdd(dst + 2, w * v.z);
    atomicAdd(dst + 3, w * v.w);
}

// ---------------- strided relu over a 256-wide layer slice ----------------
__global__ void k_relu(float* __restrict__ base, int nrows) {
    long long i = (long long)blockIdx.x * blockDim.x + threadIdx.x;
    long long total = (long long)nrows * ENT_F;
    if (i >= total) return;
    int n = (int)(i >> 8);
    int c = (int)i & 255;
    float* p = base + (size_t)n * OUT_STRIDE + c;
    float v = *p;
    *p = v > 0.0f ? v : 0.0f;
}

// ---------------- rel_proj[r] = rel_emb[r,:] . a_rel ----------------
__global__ void k_rel_proj(const float* __restrict__ rel_emb, const float* __restrict__ a_rel,
                           float* __restrict__ rel_proj, int r) {
    int i = blockIdx.x * blockDim.x + threadIdx.x;
    if (i >= r) return;
    const float* e = rel_emb + (size_t)i * NODE_F;
    float acc = 0.0f;
    for (int k = 0; k < NODE_F; ++k) acc += e[k] * a_rel[k];
    rel_proj[i] = acc;
}

// ---------------- rel_attn[sp_row[e]] += sp_val[e] * rel_proj[sp_col[e]] ----------------
__global__ void k_rel_attn(const float* __restrict__ sp_val, const int* __restrict__ sp_row,
                           const int* __restrict__ sp_col, const float* __restrict__ rel_proj,
                           float* __restrict__ rel_attn, int e) {
    int i = blockIdx.x * blockDim.x + threadIdx.x;
    if (i < e) atomicAdd(rel_attn + sp_row[i], sp_val[i] * rel_proj[sp_col[i]]);
}

// ---------------- WMMA projection: a_s/a_n = F[n,:256] . a_self/a_neigh ----------------
// One wave handles 16 nodes. D = A(16x4) x B(4x16) + C, iterated over K=256.
//   A = weights: row M=0 is a_self chunk, M=1 is a_neigh chunk, rows 2..15 zero
//       (fed branch-free from a zeroed scratch row so the inner loop has no EXEC churn).
//   B = feature tile transposed: B[k][n] = F[base+n][kk+k].
//   C(M=0, N=n) -> a_s of node base+n lives in c[0] lanes 0..15
//   C(M=1, N=n) -> a_n lives in c[1] lanes 0..15
__global__ void k_proj16(const float* __restrict__ F,        // row stride OUT_STRIDE
                         const float* __restrict__ a_self,
                         const float* __restrict__ a_neigh,
                         const float* __restrict__ zeros,    // >= ENT_F zeroed floats
                         float* __restrict__ a_s, float* __restrict__ a_n, int nNodes) {
    long long t = (long long)blockIdx.x * blockDim.x + threadIdx.x;
    int wave = (int)(t >> 5);
    int lane = (int)t & 31;
    int base = wave * 16;
    if (base >= nNodes) return;              // wave-uniform exit; EXEC all-ones inside
    int m    = lane & 15;                    // A row index / B column (node) index
    int half = lane >> 4;                    // K sub-pair selector
    const float* brow = F + (size_t)(base + m) * OUT_STRIDE + 2 * half;
#if defined(__gfx1250__) && __has_builtin(__builtin_amdgcn_wmma_f32_16x16x4_f32)
    // Branch-free per-lane A row pointer: lanes m>=2 read the zero row.
    const float* wrow = (m == 0) ? a_self : ((m == 1) ? a_neigh : zeros);
    wrow += 2 * half;
    v8f c = {0.f, 0.f, 0.f, 0.f, 0.f, 0.f, 0.f, 0.f};
    #pragma unroll 4
    for (int kk = 0; kk < ENT_F; kk += 4) {
        v2f b = *(const v2f*)(brow + kk);    // global_load_b64
        v2f a = *(const v2f*)(wrow + kk);    // global_load_b64 (zero rows broadcast-hit cache)
        c = __builtin_amdgcn_wmma_f32_16x16x4_f32(
                /*neg_a=*/false, a, /*neg_b=*/false, b,
                /*c_mod=*/(short)0, c, /*reuse_a=*/false, /*reuse_b=*/false);
    }
    if (lane < 16) {
        a_s[base + lane] = c[0];
        a_n[base + lane] = c[1];
    }
#else
    // Fallback (also what the host pass parses): lanes 0..15 do a_self dots,
    // lanes 16..31 do a_neigh dots, same launch geometry.
    (void)zeros;
    const float* w = (half == 0) ? a_self : a_neigh;
    const float* row = F + (size_t)(base + m) * OUT_STRIDE;
    float acc = 0.0f;
    for (int k = 0; k < ENT_F; ++k) acc += row[k] * w[k];
    if (half == 0) a_s[base + m] = acc;
    else           a_n[base + m] = acc;
#endif
}

// ---------------- logits + leaky relu + row max ----------------
__global__ void k_att(const float* __restrict__ rel_attn, const float* __restrict__ a_s,
                      const float* __restrict__ a_n, const int* __restrict__ arow,
                      const int* __restrict__ acol, float* __restrict__ att,
                      unsigned* __restrict__ row_max, int e) {
    int i = blockIdx.x * blockDim.x + threadIdx.x;
    if (i >= e) return;
    int r = arow[i];
    float x = rel_attn[i] + a_s[r] + a_n[acol[i]];
    x = (x >= 0.0f) ? x : 0.2f * x;          // leaky_relu slope 0.2
    att[i] = x;
    atomicMax(row_max + r, enc_f32(x));
}

// ---------------- exp(x - rowmax) and row sum ----------------
__global__ void k_expsum(float* __restrict__ att, const unsigned* __restrict__ row_max,
                         float* __restrict__ row_sum, const int* __restrict__ arow, int e) {
    int i = blockIdx.x * blockDim.x + threadIdx.x;
    if (i >= e) return;
    int r = arow[i];
    float v = expf(att[i] - dec_f32(row_max[r]));
    att[i] = v;
    atomicAdd(row_sum + r, v);
}

// ---------------- normalize (optionally emit to output tail) ----------------
__global__ void k_norm(float* __restrict__ att, const float* __restrict__ row_sum,
                       const int* __restrict__ arow, float* __restrict__ att_out, int e) {
    int i = blockIdx.x * blockDim.x + threadIdx.x;
    if (i >= e) return;
    float a = att[i] / row_sum[arow[i]];
    att[i] = a;
    if (att_out) att_out[i] = a;
}

// ---------------- weighted scatter aggregation: 64 threads per edge ----------------
__global__ void k_aggregate(const float* __restrict__ att, const int* __restrict__ arow,
                            const int* __restrict__ acol, const float* __restrict__ Fin,
                            float* __restrict__ Fout, int e) {
    long long t = (long long)blockIdx.x * blockDim.x + threadIdx.x;
    int edge = (int)(t >> 6);
    int c = ((int)t & 63) * 4;
    if (edge >= e) return;
    float w = att[edge];
    const float4 v = *(const float4*)(Fin + (size_t)acol[edge] * OUT_STRIDE + c);
    float* dst = Fout + (size_t)arow[edge] * OUT_STRIDE + c;
    atomicAdd(dst + 0, w * v.x);
    atomicAdd(dst + 1, w * v.y);
    atomicAdd(dst + 2, w * v.z);
    atomicAdd(dst + 3, w * v.w);
}

extern "C" void kernel_launch(void* const* d_in, const int* in_sizes, int n_in,
                              void* d_out, int out_size, void* d_ws, size_t ws_size,
                              hipStream_t stream) {
    const float* ent_emb = (const float*)d_in[0];
    const float* rel_emb = (const float*)d_in[1];
    const float* a_self  = (const float*)d_in[2];
    const float* a_neigh = (const float*)d_in[3];
    const float* a_rel   = (const float*)d_in[4];
    const float* sp_val  = (const float*)d_in[5];
    const int* adj_row = (const int*)d_in[6];
    const int* adj_col = (const int*)d_in[7];
    const int* sp_row  = (const int*)d_in[8];
    const int* sp_col  = (const int*)d_in[9];
    const int* rel_row = (const int*)d_in[10];
    const int* rel_col = (const int*)d_in[11];
    const int* ent_row = (const int*)d_in[12];
    const int* ent_col = (const int*)d_in[13];

    const int N  = in_sizes[0] / NODE_F;   // 100000
    const int R  = in_sizes[1] / NODE_F;   // 2000
    const int E  = in_sizes[5];            // 300000
    const int ER = in_sizes[10];           // 300000
    const int EE = in_sizes[12];           // 300000

    float* out = (float*)d_out;
    float* att_out_tail = out + (size_t)N * OUT_STRIDE;

    // workspace layout (floats): deg_rel[N] deg_ent[N] a_s[N] a_n[N]
    //                            row_max[N](uint) row_sum[N] rel_attn[E] zeros[ENT_F]
    //                            att[E] rel_proj[R]
    float* ws       = (float*)d_ws;
    float* deg_rel  = ws;
    float* deg_ent  = deg_rel + N;
    float* a_s      = deg_ent + N;
    float* a_n      = a_s + N;
    unsigned* row_max = (unsigned*)(a_n + N);
    float* row_sum  = (float*)row_max + N;
    float* rel_attn = row_sum + N;
    float* zeros    = rel_attn + E;
    float* att      = zeros + ENT_F;
    float* rel_proj = att + E;

    const int B = 256;
    auto blocks = [](long long n, int b) { return (unsigned)((n + b - 1) / b); };

    // Zero all accumulation targets (output is accumulated in place with atomics).
    k_zero<<<4096, B, 0, stream>>>(out, (long long)out_size);
    // deg, a_s/a_n, rowmax/rowsum, rel_attn, zeros
    k_zero<<<2048, B, 0, stream>>>(ws, (long long)(6LL * N + E + ENT_F));

    // Degrees, then mean aggregation directly into layer-0 slots of out.
    k_deg<<<blocks(ER, B), B, 0, stream>>>(rel_row, ER, deg_rel);
    k_deg<<<blocks(EE, B), B, 0, stream>>>(ent_row, EE, deg_ent);
    k_mean_agg<<<blocks((long long)EE * 32, B), B, 0, stream>>>(ent_row, ent_col, ent_emb, deg_ent, out + 0,      EE);
    k_mean_agg<<<blocks((long long)ER * 32, B), B, 0, stream>>>(rel_row, rel_col, rel_emb, deg_rel, out + NODE_F, ER);
    k_relu<<<blocks((long long)N * ENT_F, B), B, 0, stream>>>(out, N);

    // Relation attention term (depth-invariant).
    k_rel_proj<<<blocks(R, B), B, 0, stream>>>(rel_emb, a_rel, rel_proj, R);
    k_rel_attn<<<blocks(E, B), B, 0, stream>>>(sp_val, sp_row, sp_col, rel_proj, rel_attn, E);

    const int DEPTH = 2;
    for (int d = 0; d < DEPTH; ++d) {
        const float* Fin = out + (size_t)d * ENT_F;
        float* Fout      = out + (size_t)(d + 1) * ENT_F;
        if (d > 0)  // re-zero row_max (sentinel enc 0 == -NaN, below everything) and row_sum
            k_zero<<<blocks(2LL * N, B), B, 0, stream>>>((float*)row_max, 2LL * N);

        int waves = (N + 15) / 16;
        k_proj16<<<blocks((long long)waves * 32, B), B, 0, stream>>>(Fin, a_self, a_neigh, zeros, a_s, a_n, N);

        k_att<<<blocks(E, B), B, 0, stream>>>(rel_attn, a_s, a_n, adj_row, adj_col, att, row_max, E);
        k_expsum<<<blocks(E, B), B, 0, stream>>>(att, row_max, row_sum, adj_row, E);
        k_norm<<<blocks(E, B), B, 0, stream>>>(att, row_sum, adj_row,
                                               (d == DEPTH - 1) ? att_out_tail : (float*)nullptr, E);

        k_aggregate<<<blocks((long long)E * 64, B), B, 0, stream>>>(att, adj_row, adj_col, Fin, Fout, E);
        k_relu<<<blocks((long long)N * ENT_F, B), B, 0, stream>>>(Fout, N);
    }
}